// MSPushPullLoss_1022202216836
// MI455X (gfx1250) — compile-verified
//
#include <hip/hip_runtime.h>

#define VAR_W   1.0f
#define DIST_W  1.0f
#define M_VAR   0.1f
#define M_DIST  1.5f
#define C_INST  16
#define NBIN    17          // labels 0..16; bin 0 unused for loss
#define BATCH   16
#define NSEG    (BATCH*NBIN) // 272

// workspace layout (floats)
#define WS_SUM   0            // [3][NSEG] per-scale sums
#define WS_CNT   (3*NSEG)     // [3][NSEG] per-scale counts
#define WS_PULL  (6*NSEG)     // [3][NSEG] per-scale pull numerators
#define WS_MEAN  (9*NSEG)     // [NSEG]
#define WS_TOTC  (10*NSEG)    // [NSEG]
#define WS_TOTAL (11*NSEG)

typedef __attribute__((ext_vector_type(2))) float v2f;
typedef __attribute__((ext_vector_type(8))) float v8f;

__global__ __launch_bounds__(256) void pp_zero(float* __restrict__ ws) {
  for (int i = threadIdx.x; i < WS_TOTAL; i += 256) ws[i] = 0.0f;
}

// Pass 1: per-(batch,label) sum & count for one scale.
// Each block handles 4096 contiguous pixels (divides every N_s, so one batch per block).
__global__ __launch_bounds__(256) void pp_pass1(const float* __restrict__ f,
                                                const int*   __restrict__ g,
                                                int Ns,
                                                float* __restrict__ sum_s,
                                                float* __restrict__ cnt_s) {
  __shared__ float lsum[NBIN];
  __shared__ float lcnt[NBIN];
  const int t = threadIdx.x;
  if (t < NBIN) { lsum[t] = 0.0f; lcnt[t] = 0.0f; }
  __syncthreads();

  const int base = blockIdx.x * 4096;
  const int b = base / Ns;
  const float4* f4 = (const float4*)(f + base);
  const int4*   g4 = (const int4*)(g + base);

#pragma unroll
  for (int it = 0; it < 4; ++it) {
    if (it < 3) { __builtin_prefetch(&f4[(it+1)*256 + t]); __builtin_prefetch(&g4[(it+1)*256 + t]); }
    float4 v = f4[it*256 + t];
    int4   l = g4[it*256 + t];
#define ACC(vv, ll) { int lab = (ll); if ((unsigned)(lab - 1) < 16u) { \
      atomicAdd(&lsum[lab], (vv)); atomicAdd(&lcnt[lab], 1.0f); } }
    ACC(v.x, l.x) ACC(v.y, l.y) ACC(v.z, l.z) ACC(v.w, l.w)
#undef ACC
  }
  __syncthreads();
  if (t >= 1 && t < NBIN && lcnt[t] > 0.0f) {
    atomicAdd(&sum_s[b*NBIN + t], lsum[t]);
    atomicAdd(&cnt_s[b*NBIN + t], lcnt[t]);
  }
}

// Per-segment mean over all scales + total counts.
__global__ __launch_bounds__(512) void pp_mean(float* __restrict__ ws) {
  const int i = threadIdx.x;
  if (i < NSEG) {
    float s = ws[WS_SUM + i] + ws[WS_SUM + NSEG + i] + ws[WS_SUM + 2*NSEG + i];
    float c = ws[WS_CNT + i] + ws[WS_CNT + NSEG + i] + ws[WS_CNT + 2*NSEG + i];
    ws[WS_MEAN + i] = s / fmaxf(c, 1.0f);
    ws[WS_TOTC + i] = c;
  }
}

// Pass 2: per-(batch,label) sum of clamp(|x-mean| - M_VAR, 0)^2 for one scale.
__global__ __launch_bounds__(256) void pp_pass2(const float* __restrict__ f,
                                                const int*   __restrict__ g,
                                                int Ns,
                                                const float* __restrict__ mean,
                                                float* __restrict__ pull_s) {
  __shared__ float lmean[NBIN];
  __shared__ float lpull[NBIN];
  const int t = threadIdx.x;
  const int base = blockIdx.x * 4096;
  const int b = base / Ns;
  if (t < NBIN) { lmean[t] = mean[b*NBIN + t]; lpull[t] = 0.0f; }
  __syncthreads();

  const float4* f4 = (const float4*)(f + base);
  const int4*   g4 = (const int4*)(g + base);

#pragma unroll
  for (int it = 0; it < 4; ++it) {
    if (it < 3) { __builtin_prefetch(&f4[(it+1)*256 + t]); __builtin_prefetch(&g4[(it+1)*256 + t]); }
    float4 v = f4[it*256 + t];
    int4   l = g4[it*256 + t];
#define ACC(vv, ll) { int lab = (ll); if ((unsigned)(lab - 1) < 16u) { \
      float d = fmaxf(fabsf((vv) - lmean[lab]) - M_VAR, 0.0f); \
      atomicAdd(&lpull[lab], d * d); } }
    ACC(v.x, l.x) ACC(v.y, l.y) ACC(v.z, l.z) ACC(v.w, l.w)
#undef ACC
  }
  __syncthreads();
  if (t >= 1 && t < NBIN && lpull[t] != 0.0f) {
    atomicAdd(&pull_s[b*NBIN + t], lpull[t]);
  }
}

// Finalize: pull normalization + push loss. One block, 16 waves; wave w handles
// batch w, computing the 16x16 all-pairs (m_i - m_j) matrix exactly in f32 via
// V_WMMA_F32_16X16X4_F32 with A=[m|1|0|0], B=[1; -m; 0; 0].
__global__ __launch_bounds__(512) void pp_finalize(const float* __restrict__ ws,
                                                   float* __restrict__ out) {
  __shared__ float acc[4]; // pull_sum, pull_cnt, push_sum, push_cnt
  const int t = threadIdx.x;
  if (t < 4) acc[t] = 0.0f;
  __syncthreads();

  // ---- pull loss: one thread per (batch, instance) ----
  if (t < BATCH * C_INST) {
    const int seg = (t >> 4) * NBIN + 1 + (t & 15);
    float ps = 0.0f;
#pragma unroll
    for (int s = 0; s < 3; ++s) {
      float sc = ws[WS_CNT  + s*NSEG + seg];
      float sv = ws[WS_PULL + s*NSEG + seg];
      ps += (sc > 0.0f) ? (sv / sc) : 0.0f;
    }
    if (ws[WS_TOTC + seg] > 0.0f) {
      atomicAdd(&acc[0], ps);
      atomicAdd(&acc[1], 1.0f);
    }
  }

  // ---- push loss via WMMA (branch-free per wave; EXEC all-ones) ----
  const int wave = t >> 5;
  const int lane = t & 31;
  const int b    = wave;              // 16 waves == 16 batches
  const int j    = lane & 15;
  const float m_l = ws[WS_MEAN + b*NBIN + 1 + j];
  const float c_l = ws[WS_TOTC + b*NBIN + 1 + j];
  const unsigned pmask = (unsigned)__ballot(c_l > 0.0f) & 0xFFFFu; // bit i = present[b][i]

  const bool lo = (lane < 16);
  // A 16x4 layout: lanes 0-15 hold row M=lane, VGPR0->K=0, VGPR1->K=1;
  // lanes 16-31 hold K=2,3 (zeros here).
  v2f a, bb;
  a[0]  = lo ?  m_l : 0.0f;  // A[i,0] = m_i
  a[1]  = lo ? 1.0f : 0.0f;  // A[i,1] = 1
  bb[0] = lo ? 1.0f : 0.0f;  // B[0,j] = 1
  bb[1] = lo ? -m_l : 0.0f;  // B[1,j] = -m_j
  v8f cz = {};
  v8f d = __builtin_amdgcn_wmma_f32_16x16x4_f32(
      /*neg_a=*/false, a, /*neg_b=*/false, bb,
      /*c_mod=*/(short)0, cz, /*reuse_a=*/false, /*reuse_b=*/false);

  // D layout: lanes 0-15: VGPR r -> (M=r, N=lane); lanes 16-31: (M=r+8, N=lane-16)
  float psum = 0.0f, pcnt = 0.0f;
#pragma unroll
  for (int r = 0; r < 8; ++r) {
    const int i = lo ? r : (r + 8);
    float pv = fmaxf(2.0f * M_DIST - fabsf(d[r]), 0.0f);
    pv *= pv;
    const bool pm = ((pmask >> i) & 1u) && ((pmask >> j) & 1u) && (i != j);
    psum += pm ? pv : 0.0f;
    pcnt += pm ? 1.0f : 0.0f;
  }
  atomicAdd(&acc[2], psum);
  atomicAdd(&acc[3], pcnt);
  __syncthreads();

  if (t == 0) {
    const float pull = acc[0] / fmaxf(acc[1], 1.0f) * VAR_W;
    const float push = acc[2] / fmaxf(acc[3], 1.0f) * DIST_W;
    out[0] = push + pull;
  }
}

extern "C" void kernel_launch(void* const* d_in, const int* in_sizes, int n_in,
                              void* d_out, int out_size, void* d_ws, size_t ws_size,
                              hipStream_t stream) {
  // setup_inputs() order: featmap0, gt0, featmap1, gt1, featmap2, gt2
  const float* f0 = (const float*)d_in[0];
  const int*   g0 = (const int*)  d_in[1];
  const float* f1 = (const float*)d_in[2];
  const int*   g1 = (const int*)  d_in[3];
  const float* f2 = (const float*)d_in[4];
  const int*   g2 = (const int*)  d_in[5];
  float* ws  = (float*)d_ws;
  float* out = (float*)d_out;

  const int n0 = in_sizes[0], n1 = in_sizes[2], n2 = in_sizes[4];
  const int Ns0 = n0 / BATCH, Ns1 = n1 / BATCH, Ns2 = n2 / BATCH;

  pp_zero<<<1, 256, 0, stream>>>(ws);

  pp_pass1<<<n0 / 4096, 256, 0, stream>>>(f0, g0, Ns0, ws + WS_SUM + 0*NSEG, ws + WS_CNT + 0*NSEG);
  pp_pass1<<<n1 / 4096, 256, 0, stream>>>(f1, g1, Ns1, ws + WS_SUM + 1*NSEG, ws + WS_CNT + 1*NSEG);
  pp_pass1<<<n2 / 4096, 256, 0, stream>>>(f2, g2, Ns2, ws + WS_SUM + 2*NSEG, ws + WS_CNT + 2*NSEG);

  pp_mean<<<1, 512, 0, stream>>>(ws);

  pp_pass2<<<n0 / 4096, 256, 0, stream>>>(f0, g0, Ns0, ws + WS_MEAN, ws + WS_PULL + 0*NSEG);
  pp_pass2<<<n1 / 4096, 256, 0, stream>>>(f1, g1, Ns1, ws + WS_MEAN, ws + WS_PULL + 1*NSEG);
  pp_pass2<<<n2 / 4096, 256, 0, stream>>>(f2, g2, Ns2, ws + WS_MEAN, ws + WS_PULL + 2*NSEG);

  pp_finalize<<<1, 512, 0, stream>>>(ws, out);
}